// GCN_MPML_Learned_3624952397851
// MI455X (gfx1250) — compile-verified
//
#include <hip/hip_runtime.h>
#include <cstdint>

// GCN_MPML_Learned: persistence-landscape gradient matrices.
// S=2000 filtration pts, P=2500 sample pts, R=5 bars, 2 bar sets.
// Output ~400MB of ±1/0 f32 -> store-bandwidth bound (>=17us @ 23.3TB/s).
// No matmul structure -> WMMA not applicable. CDNA5 paths used instead:
//   * Tensor Data Mover (tensor_load_to_lds + s_wait_tensorcnt) to stage the
//     16KB filtration operand into LDS (reused 12,500x per byte),
//   * wave32 __shfl_xor butterfly for the any()-reductions,
//   * non-temporal 16B stores for the streaming gm output.

#define BLK  256
#define SMAX 2048   // S=2000 <= 2048

typedef float        v2f __attribute__((ext_vector_type(2)));
typedef float        v4f __attribute__((ext_vector_type(4)));
typedef unsigned int v4u __attribute__((ext_vector_type(4)));
typedef int          v4i __attribute__((ext_vector_type(4)));
typedef int          v8i __attribute__((ext_vector_type(8)));

__global__ __launch_bounds__(BLK) void mpml_grad_kernel(
    const float* __restrict__ filt,    // (S,2)
    const float* __restrict__ bars0,   // (P,R)
    const float* __restrict__ bars1,   // (P,R)
    const float* __restrict__ pts,     // (P,2)
    float* __restrict__ outbuf,        // concatenated outputs
    int S, int P, int R)
{
  __shared__ v2f rawf[SMAX];           // {fx, fy}
  __shared__ v4f sf[SMAX];             // {fx, fy, tol_x, tol_y}
  __shared__ unsigned accF[16];        // per-(h,r) any-flags

  const int tid = threadIdx.x;
  const int p   = blockIdx.x;

  if (tid < 2 * R) accF[tid] = 0u;

  // ---- Stage filtration (2S f32, 16KB) into LDS via the Tensor Data Mover.
#if defined(__gfx1250__) && __has_builtin(__builtin_amdgcn_tensor_load_to_lds)
  if (tid < 32) {  // exactly wave 0 issues the DMA
    const unsigned lds_addr = (unsigned)(uintptr_t)(&rawf[0]); // low 32b of LDS aperture = LDS byte offset
    const unsigned long long ga = (unsigned long long)(uintptr_t)filt;
    const unsigned n = (unsigned)(2 * S);  // elements, fits 16 bits (<=4096)

    // D# group 0: count=1 | lds_addr | global_addr[56:0] | type=2
    v4u g0;
    g0.x = 1u;
    g0.y = lds_addr;
    g0.z = (unsigned)(ga & 0xFFFFFFFFull);
    g0.w = (unsigned)((ga >> 32) & 0x01FFFFFFull) | (2u << 30);

    // D# group 1: data_size=4B, tensor = n x 1, tile = n x 1, stride0 = n
    v8i g1;
    g1[0] = (int)(2u << 16);                               // data_size=2 (4B), mask/flags=0
    g1[1] = (int)((n & 0xFFFFu) << 16);                    // tensor_dim0[15:0]
    g1[2] = (int)(((n >> 16) & 0xFFFFu) | (1u << 16));     // tensor_dim0[31:16] | tensor_dim1=1
    g1[3] = (int)((n & 0xFFFFu) << 16);                    // tensor_dim1 hi=0 | tile_dim0=n
    g1[4] = 1;                                             // tile_dim1=1, tile_dim2=0
    g1[5] = (int)n;                                        // tensor_dim0_stride lo
    g1[6] = 0;
    g1[7] = 0;

    v4i z4 = {0, 0, 0, 0};
#if defined(__clang_major__) && __clang_major__ >= 23
    v8i z8 = {0, 0, 0, 0, 0, 0, 0, 0};
    __builtin_amdgcn_tensor_load_to_lds(g0, g1, z4, z4, z8, 0);
#else
    __builtin_amdgcn_tensor_load_to_lds(g0, g1, z4, z4, 0);
#endif
    __builtin_amdgcn_s_wait_tensorcnt(0);
  }
#else
  for (int s = tid; s < S; s += BLK) rawf[s] = ((const v2f*)filt)[s];
#endif
  __syncthreads();

  // Precompute per-s {fx, fy, tol_x, tol_y} once (hoists 4 ops x10 reuses).
  // Op order matches jnp: ATOL + RTOL*|f| as separate rounded mul then add.
  for (int s = tid; s < S; s += BLK) {
    v2f f = rawf[s];
    v4f e;
    e.x = f.x;
    e.y = f.y;
    e.z = __fadd_rn(0.01f, __fmul_rn(1e-5f, fabsf(f.x)));
    e.w = __fadd_rn(0.01f, __fmul_rn(1e-5f, fabsf(f.y)));
    sf[s] = e;
  }
  __syncthreads();

  const float ptx  = pts[2 * p + 0];
  const float pty  = pts[2 * p + 1];
  const float psum = __fadd_rn(ptx, pty);

  const size_t PR   = (size_t)P * (size_t)R;
  const size_t S2   = (size_t)2 * (size_t)S;
  const size_t GMSZ = (size_t)R * (size_t)P * S2;
  const int    SH   = S >> 1;           // number of s-pairs (S=2000 -> 1000)
  // layout: out[2PR] | gm0[GMSZ] | gc0[2PR] | gm1[GMSZ] | gc1[2PR]

  for (int h = 0; h < 2; ++h) {
    const float* __restrict__ bars = h ? bars1 : bars0;
    float* __restrict__ gm = outbuf + 2 * PR + (size_t)h * (GMSZ + 2 * PR);

    for (int r = 0; r < R; ++r) {
      const float scale = __fadd_rn(bars[(size_t)p * R + r], 0.01f);
      float lx[5], ly[5];
#pragma unroll
      for (int k = 0; k < 5; ++k) {          // shift = -2..2, line = shift*scale + pt
        const float sh = (float)(k - 2);
        lx[k] = __fadd_rn(__fmul_rn(sh, scale), ptx);
        ly[k] = __fadd_rn(__fmul_rn(sh, scale), pty);
      }
      const float lastx = lx[4];
      const float lasty = ly[4];
      unsigned flags = 0u;  // bit0: any upper_x, bit1: lower_x, bit2: upper_y, bit3: lower_y

      v4f* __restrict__ row4 = (v4f*)(gm + ((size_t)r * P + p) * S2);

      auto evalPt = [&](int s) -> v2f {
        const v4f f = sf[s];
        bool cx = false, cy = false;
#pragma unroll
        for (int k = 0; k < 5; ++k) {
          cx = cx || (fabsf(__fsub_rn(lx[k], f.x)) <= f.z);
          cy = cy || (fabsf(__fsub_rn(ly[k], f.y)) <= f.w);
        }
        const float fs   = __fadd_rn(f.x, f.y);
        const bool below = fs < psum;
        const bool above = fs > psum;
        const bool condx = cx && (f.y <= lasty);
        const bool condy = cy && (f.x <= lastx);
        const bool ux = condx && above;
        const bool dx = condx && below;
        const bool uy = condy && above;
        const bool dy = condy && below;
        flags |= (ux ? 1u : 0u) | (dx ? 2u : 0u) | (uy ? 4u : 0u) | (dy ? 8u : 0u);
        v2f g;
        g.x = ux ? 1.0f : (dx ? -1.0f : 0.0f);
        g.y = uy ? 1.0f : (dy ? -1.0f : 0.0f);
        return g;
      };

      // Two s-values per thread -> one non-temporal 16B store, no tail branch
      // in the hot loop (S even covers everything; odd tail handled below).
      for (int si = tid; si < SH; si += BLK) {
        const int s0 = 2 * si;
        const v2f e0 = evalPt(s0);
        const v2f e1 = evalPt(s0 + 1);
        v4f g;
        g.x = e0.x; g.y = e0.y; g.z = e1.x; g.w = e1.y;
        __builtin_nontemporal_store(g, &row4[si]);
      }
      if ((S & 1) && tid == 0) {  // robustness only; never taken for S=2000
        const v2f e = evalPt(S - 1);
        __builtin_nontemporal_store(e, (v2f*)row4 + (S - 1));
      }

      // wave32 butterfly-OR, then one LDS atomic per wave
#pragma unroll
      for (int off = 16; off > 0; off >>= 1)
        flags |= __shfl_xor(flags, off, 32);
      if ((tid & 31) == 0 && flags) atomicOr(&accF[h * R + r], flags);
    }
  }

  __syncthreads();
  if (tid < 2 * R) {
    const int h = tid / R;
    const int r = tid % R;
    const unsigned fl = accF[tid];
    const float cx = (fl & 1u) ? -1.0f : ((fl & 2u) ? 1.0f : 0.0f);
    const float cy = (fl & 4u) ? -1.0f : ((fl & 8u) ? 1.0f : 0.0f);
    float* gc = outbuf + 2 * PR + GMSZ + (size_t)h * (GMSZ + 2 * PR);
    const size_t gi = ((size_t)r * P + p) * 2;
    gc[gi + 0] = cx;
    gc[gi + 1] = cy;
    // out = stack([bars_h0, bars_h1])
    const float* bars = h ? bars1 : bars0;
    outbuf[(size_t)h * PR + (size_t)p * R + r] = bars[(size_t)p * R + r];
  }
}

extern "C" void kernel_launch(void* const* d_in, const int* in_sizes, int n_in,
                              void* d_out, int out_size, void* d_ws, size_t ws_size,
                              hipStream_t stream) {
  const float* filt = (const float*)d_in[0];   // (S,2)
  const float* b0   = (const float*)d_in[1];   // (P,R)
  const float* b1   = (const float*)d_in[2];   // (P,R)
  const float* pts  = (const float*)d_in[3];   // (P,2)

  const int S = in_sizes[0] / 2;
  const int P = in_sizes[3] / 2;
  const int R = in_sizes[1] / P;

  mpml_grad_kernel<<<P, BLK, 0, stream>>>(filt, b0, b1, pts, (float*)d_out, S, P, R);
}